// WisePooling_64424509440374
// MI455X (gfx1250) — compile-verified
//
#include <hip/hip_runtime.h>

// Segment-mean pooling on MI455X (gfx1250, wave32).
// out[s, :] = mean(x[start_s .. end_s, :], axis=0) + EPS
// Row-reduction done on the matrix pipe: V_WMMA_F32_16X16X4_F32 with A = ones.

typedef float v2f __attribute__((ext_vector_type(2)));
typedef float v8f __attribute__((ext_vector_type(8)));

#define D_DIM 512
#define EPS_F 0.006f

__global__ __launch_bounds__(256)
void seg_mean_wmma(const float* __restrict__ x,
                   const int*   __restrict__ graph,
                   float*       __restrict__ out)
{
    const int s     = blockIdx.x;
    const int start = graph[2 * s + 0];
    const int end   = graph[2 * s + 1];
    const int count = end - start + 1;

    const int lane = threadIdx.x & 31;
    const int wave = threadIdx.x >> 5;   // 0..7
    const int half = lane >> 4;          // 0 or 1 (which K-row this lane carries)
    const int col  = lane & 15;          // N position within a 16-col slab

    // 4 accumulators per wave -> 4 slabs of 16 columns = 64 cols/wave, 512 total.
    v8f acc0 = {}, acc1 = {}, acc2 = {}, acc3 = {};

    // A-matrix: 16x4 all-ones (2 VGPRs/lane, every element 1.0).
    v2f ones;
    ones[0] = 1.0f;
    ones[1] = 1.0f;

    const float* xrow = x + (size_t)start * D_DIM;

    const int c0 = (wave +  0) * 16 + col;
    const int c1 = (wave +  8) * 16 + col;
    const int c2 = (wave + 16) * 16 + col;
    const int c3 = (wave + 24) * 16 + col;

    // Main loop: 4 rows per iteration, unconditional coalesced B32 loads.
    const int full = count & ~3;
    for (int r = 0; r < full; r += 4) {
        const size_t ra = (size_t)(r + half)     * D_DIM;  // rows r, r+1
        const size_t rb = (size_t)(r + 2 + half) * D_DIM;  // rows r+2, r+3

        v2f b0; b0[0] = xrow[ra + c0]; b0[1] = xrow[rb + c0];
        v2f b1; b1[0] = xrow[ra + c1]; b1[1] = xrow[rb + c1];
        v2f b2; b2[0] = xrow[ra + c2]; b2[1] = xrow[rb + c2];
        v2f b3; b3[0] = xrow[ra + c3]; b3[1] = xrow[rb + c3];

        acc0 = __builtin_amdgcn_wmma_f32_16x16x4_f32(false, ones, false, b0,
                                                     (short)0, acc0, false, false);
        acc1 = __builtin_amdgcn_wmma_f32_16x16x4_f32(false, ones, false, b1,
                                                     (short)0, acc1, false, false);
        acc2 = __builtin_amdgcn_wmma_f32_16x16x4_f32(false, ones, false, b2,
                                                     (short)0, acc2, false, false);
        acc3 = __builtin_amdgcn_wmma_f32_16x16x4_f32(false, ones, false, b3,
                                                     (short)0, acc3, false, false);
    }

    // Tail quad: zero-fill rows past the segment (select keeps EXEC all-ones
    // for the WMMA itself).
    if (full < count) {
        const int r  = full;
        const int ra = r + half;
        const int rb = r + 2 + half;
        const bool va = ra < count;
        const bool vb = rb < count;
        const size_t oa = (size_t)ra * D_DIM;
        const size_t ob = (size_t)rb * D_DIM;

        v2f b0; b0[0] = va ? xrow[oa + c0] : 0.0f; b0[1] = vb ? xrow[ob + c0] : 0.0f;
        v2f b1; b1[0] = va ? xrow[oa + c1] : 0.0f; b1[1] = vb ? xrow[ob + c1] : 0.0f;
        v2f b2; b2[0] = va ? xrow[oa + c2] : 0.0f; b2[1] = vb ? xrow[ob + c2] : 0.0f;
        v2f b3; b3[0] = va ? xrow[oa + c3] : 0.0f; b3[1] = vb ? xrow[ob + c3] : 0.0f;

        acc0 = __builtin_amdgcn_wmma_f32_16x16x4_f32(false, ones, false, b0,
                                                     (short)0, acc0, false, false);
        acc1 = __builtin_amdgcn_wmma_f32_16x16x4_f32(false, ones, false, b1,
                                                     (short)0, acc1, false, false);
        acc2 = __builtin_amdgcn_wmma_f32_16x16x4_f32(false, ones, false, b2,
                                                     (short)0, acc2, false, false);
        acc3 = __builtin_amdgcn_wmma_f32_16x16x4_f32(false, ones, false, b3,
                                                     (short)0, acc3, false, false);
    }

    // D layout: VGPR0 lanes 0..15 hold (M=0, N=lane). A==ones makes every M row
    // identical, so component 0 on lanes 0..15 is the column sum.
    const float inv = 1.0f / (float)count;
    if (lane < 16) {
        float* o = out + (size_t)s * D_DIM;
        o[c0] = acc0[0] * inv + EPS_F;
        o[c1] = acc1[0] * inv + EPS_F;
        o[c2] = acc2[0] * inv + EPS_F;
        o[c3] = acc3[0] * inv + EPS_F;
    }
}

extern "C" void kernel_launch(void* const* d_in, const int* in_sizes, int n_in,
                              void* d_out, int out_size, void* d_ws, size_t ws_size,
                              hipStream_t stream) {
    const float* x     = (const float*)d_in[0];
    const int*   graph = (const int*)d_in[1];
    float*       out   = (float*)d_out;

    const int S = in_sizes[1] / 2;  // graph is (S, 2) int32

    dim3 grid(S);
    dim3 block(256);
    seg_mean_wmma<<<grid, block, 0, stream>>>(x, graph, out);
}